// StructureGNN_15341623181529
// MI455X (gfx1250) — compile-verified
//
#include <hip/hip_runtime.h>

typedef __attribute__((ext_vector_type(2))) float v2f;
typedef __attribute__((ext_vector_type(8))) float v8f;

#define N_NODES 50000
#define N_EDGES 800000
#define N_GRAPHS 256

// ---------------------------------------------------------------------------
// Utility: zero a float buffer
// ---------------------------------------------------------------------------
__global__ void zero_f32(float* __restrict__ p, int n) {
  int i = blockIdx.x * blockDim.x + threadIdx.x;
  if (i < n) p[i] = 0.0f;
}

// ---------------------------------------------------------------------------
// Degree count over dst (self-loops folded in later as +1)
// ---------------------------------------------------------------------------
__global__ void degree_count(const int* __restrict__ dst, float* __restrict__ deg, int e) {
  int i = blockIdx.x * blockDim.x + threadIdx.x;
  if (i < e) atomicAdd(&deg[dst[i]], 1.0f);
}

__global__ void deg_to_dinv(float* __restrict__ deg, int n) {
  int i = blockIdx.x * blockDim.x + threadIdx.x;
  if (i < n) deg[i] = rsqrtf(deg[i] + 1.0f);  // +1 for self-loop
}

// ---------------------------------------------------------------------------
// WMMA fp32 GEMM: out[M,64] = A[M,K] @ W[K,64] (+ bias).  K % 4 == 0.
// One wave computes a 16-row x 64-col strip via 4 accumulator tiles.
// V_WMMA_F32_16X16X4_F32:
//   A 16x4 : lanes 0-15 -> K={0,1}, lanes 16-31 -> K={2,3}, row M = lane&15
//   B 4x16 : lanes 0-15 -> K={0,1}, lanes 16-31 -> K={2,3}, col N = lane&15
//   C 16x16: VGPR v -> M = v + 8*(lane>=16), N = lane&15
// ---------------------------------------------------------------------------
__global__ void gemm_n64_wmma(const float* __restrict__ A, const float* __restrict__ W,
                              const float* __restrict__ bias, float* __restrict__ out,
                              int mTiles, int K) {
  const int lane = threadIdx.x & 31;
  const int wave = threadIdx.x >> 5;
  const int tile = blockIdx.x * (blockDim.x >> 5) + wave;
  if (tile >= mTiles) return;  // wave-uniform: EXEC stays all-ones for WMMA

  const int half = lane >> 4;    // 0 for lanes 0-15, 1 for lanes 16-31
  const int l15  = lane & 15;
  const int rowA = tile * 16 + l15;
  const int koff = half * 2;     // lanes 16-31 carry K+2, K+3

  v8f acc0 = {}, acc1 = {}, acc2 = {}, acc3 = {};
  const int kSteps = K >> 2;
  for (int kk = 0; kk < kSteps; ++kk) {
    const int kb = kk * 4 + koff;
    v2f a;
    a.x = A[(size_t)rowA * K + kb];
    a.y = A[(size_t)rowA * K + kb + 1];
    const float* w0 = W + (size_t)kb * 64;
    const float* w1 = w0 + 64;
    v2f b0, b1, b2, b3;
    b0.x = w0[l15];      b0.y = w1[l15];
    b1.x = w0[16 + l15]; b1.y = w1[16 + l15];
    b2.x = w0[32 + l15]; b2.y = w1[32 + l15];
    b3.x = w0[48 + l15]; b3.y = w1[48 + l15];
    acc0 = __builtin_amdgcn_wmma_f32_16x16x4_f32(false, a, false, b0, (short)0, acc0, false, false);
    acc1 = __builtin_amdgcn_wmma_f32_16x16x4_f32(false, a, false, b1, (short)0, acc1, false, false);
    acc2 = __builtin_amdgcn_wmma_f32_16x16x4_f32(false, a, false, b2, (short)0, acc2, false, false);
    acc3 = __builtin_amdgcn_wmma_f32_16x16x4_f32(false, a, false, b3, (short)0, acc3, false, false);
  }

  float bv0 = 0.f, bv1 = 0.f, bv2 = 0.f, bv3 = 0.f;
  if (bias) { bv0 = bias[l15]; bv1 = bias[16 + l15]; bv2 = bias[32 + l15]; bv3 = bias[48 + l15]; }

  const int mBase = tile * 16 + half * 8;
#pragma unroll
  for (int v = 0; v < 8; ++v) {
    float* o = out + (size_t)(mBase + v) * 64;
    o[l15]      = acc0[v] + bv0;
    o[16 + l15] = acc1[v] + bv1;
    o[32 + l15] = acc2[v] + bv2;
    o[48 + l15] = acc3[v] + bv3;
  }
}

// ---------------------------------------------------------------------------
// Edge scatter: agg[dst] += dinv[src]*dinv[dst] * t[src]  (64 feats/edge)
// 16 threads per edge, float4 gather, 4 fp32 atomics per thread (L2-resident).
// ---------------------------------------------------------------------------
__global__ void scatter_edges(const int* __restrict__ src, const int* __restrict__ dst,
                              const float* __restrict__ dinv, const float* __restrict__ t,
                              float* __restrict__ agg, int nEdges) {
  int tid = blockIdx.x * blockDim.x + threadIdx.x;
  int e = tid >> 4;
  if (e >= nEdges) return;
  int fl = tid & 15;           // feature group: 4 floats each
  int s = src[e], d = dst[e];
  float nrm = dinv[s] * dinv[d];
  const float4 v = ((const float4*)(t + (size_t)s * 64))[fl];
  float* ap = agg + (size_t)d * 64 + fl * 4;
  atomicAdd(ap + 0, nrm * v.x);
  atomicAdd(ap + 1, nrm * v.y);
  atomicAdd(ap + 2, nrm * v.z);
  atomicAdd(ap + 3, nrm * v.w);
}

// ---------------------------------------------------------------------------
// Epilogue: h = relu(agg + dinv[i]^2 * t[i] + b)   (self-loop term fused)
// ---------------------------------------------------------------------------
__global__ void finalize_layer(const float* __restrict__ t, const float* __restrict__ dinv,
                               const float* __restrict__ b, float* __restrict__ agg, int nNodes) {
  int idx = blockIdx.x * blockDim.x + threadIdx.x;
  if (idx >= nNodes * 64) return;
  int i = idx >> 6, f = idx & 63;
  float di = dinv[i];
  float v = agg[idx] + di * di * t[idx] + b[f];
  agg[idx] = fmaxf(v, 0.0f);
}

// ---------------------------------------------------------------------------
// Mean pool: atomic sums per graph id + counts
// ---------------------------------------------------------------------------
__global__ void pool_sum(const float* __restrict__ h, const int* __restrict__ batch,
                         float* __restrict__ sums, float* __restrict__ cnts, int nNodes) {
  int t = blockIdx.x * blockDim.x + threadIdx.x;
  if (t >= nNodes * 64) return;
  int i = t >> 6, f = t & 63;
  int g = batch[i];
  atomicAdd(&sums[(size_t)g * 64 + f], h[t]);
  if (f == 0) atomicAdd(&cnts[g], 1.0f);
}

__global__ void pool_div(const float* __restrict__ sums, const float* __restrict__ cnts,
                         float* __restrict__ pooled) {
  int t = blockIdx.x * blockDim.x + threadIdx.x;
  if (t >= N_GRAPHS * 64) return;
  pooled[t] = sums[t] / fmaxf(cnts[t >> 6], 1.0f);
}

// ---------------------------------------------------------------------------
extern "C" void kernel_launch(void* const* d_in, const int* in_sizes, int n_in,
                              void* d_out, int out_size, void* d_ws, size_t ws_size,
                              hipStream_t stream) {
  const float* x    = (const float*)d_in[0];
  const int*   edge = (const int*)d_in[1];
  const int*   batch= (const int*)d_in[2];
  const float* W1   = (const float*)d_in[3];
  const float* b1   = (const float*)d_in[4];
  const float* W2   = (const float*)d_in[5];
  const float* b2   = (const float*)d_in[6];
  const float* Wout = (const float*)d_in[7];
  const float* bout = (const float*)d_in[8];
  float* out = (float*)d_out;

  const int* src = edge;
  const int* dst = edge + N_EDGES;

  // Workspace layout (floats): ~26 MB total
  float* ws    = (float*)d_ws;
  float* bufA  = ws;                                  // 50000*64  (t1 / t2)
  float* bufB  = bufA + (size_t)N_NODES * 64;         // 50000*64  (agg / h)
  float* dinv  = bufB + (size_t)N_NODES * 64;         // 50000
  float* sums  = dinv + N_NODES;                      // 256*64
  float* cnts  = sums + (size_t)N_GRAPHS * 64;        // 256
  float* pooled= cnts + N_GRAPHS;                     // 256*64

  const int BLK = 256;
  const int nodeFeat = N_NODES * 64;                  // 3,200,000
  const int gNodeFeat = (nodeFeat + BLK - 1) / BLK;   // 12500
  const int gScatter  = (N_EDGES * 16 + BLK - 1) / BLK; // 50000
  const int mTiles = N_NODES / 16;                    // 3125 (exact)
  const int gGemm  = (mTiles + 7) / 8;                // 8 waves/block

  // --- normalization: deg (incl. self-loop) -> dinv ---
  zero_f32<<<(N_NODES + BLK - 1) / BLK, BLK, 0, stream>>>(dinv, N_NODES);
  degree_count<<<(N_EDGES + BLK - 1) / BLK, BLK, 0, stream>>>(dst, dinv, N_EDGES);
  deg_to_dinv<<<(N_NODES + BLK - 1) / BLK, BLK, 0, stream>>>(dinv, N_NODES);

  // --- layer 1: t1 = x@W1 ; agg = scatter ; h1 = relu(agg + dinv^2*t1 + b1) ---
  gemm_n64_wmma<<<gGemm, BLK, 0, stream>>>(x, W1, nullptr, bufA, mTiles, 20);
  zero_f32<<<gNodeFeat, BLK, 0, stream>>>(bufB, nodeFeat);
  scatter_edges<<<gScatter, BLK, 0, stream>>>(src, dst, dinv, bufA, bufB, N_EDGES);
  finalize_layer<<<gNodeFeat, BLK, 0, stream>>>(bufA, dinv, b1, bufB, N_NODES);

  // --- layer 2: t2 = h1@W2 ; agg ; h2 ---
  gemm_n64_wmma<<<gGemm, BLK, 0, stream>>>(bufB, W2, nullptr, bufA, mTiles, 64);
  zero_f32<<<gNodeFeat, BLK, 0, stream>>>(bufB, nodeFeat);
  scatter_edges<<<gScatter, BLK, 0, stream>>>(src, dst, dinv, bufA, bufB, N_EDGES);
  finalize_layer<<<gNodeFeat, BLK, 0, stream>>>(bufA, dinv, b2, bufB, N_NODES);

  // --- global mean pool ---
  zero_f32<<<(N_GRAPHS * 64 + N_GRAPHS + BLK - 1) / BLK, BLK, 0, stream>>>(sums, N_GRAPHS * 64 + N_GRAPHS);
  pool_sum<<<gNodeFeat, BLK, 0, stream>>>(bufB, batch, sums, cnts, N_NODES);
  pool_div<<<(N_GRAPHS * 64 + BLK - 1) / BLK, BLK, 0, stream>>>(sums, cnts, pooled);

  // --- head: out = pooled@Wout + bout  (256x64 @ 64x64) ---
  gemm_n64_wmma<<<(N_GRAPHS / 16 + 7) / 8, BLK, 0, stream>>>(pooled, Wout, bout, out, N_GRAPHS / 16, 64);
}